// GeneralizedGraphDiffusion_17841294147718
// MI455X (gfx1250) — compile-verified
//
#include <hip/hip_runtime.h>

typedef float v2f __attribute__((ext_vector_type(2)));
typedef float v4f __attribute__((ext_vector_type(4)));
typedef float v8f __attribute__((ext_vector_type(8)));

namespace {
constexpr int Nn   = 8192;  // graph size
constexpr int DIN  = 128;
constexpr int DOUT = 128;
constexpr int TJ   = 64;    // q column-strip width per iteration
constexpr int QSTR = TJ + 4;    // padded q row stride (68: stride%64==4 -> conflict-free)
constexpr int HSTR = DIN + 4;   // padded h row stride (132)
}

__global__ __launch_bounds__(256)
void ggd_fused_kernel(const float* __restrict__ theta,
                      const float* __restrict__ T,      // [4][N][N]  streamed once (NT)
                      const float* __restrict__ x,      // [N][128]   L2-resident, reused
                      const float* __restrict__ a,      // [N][N]     streamed once (NT)
                      const float* __restrict__ alpha,  // [128]
                      const float* __restrict__ W,      // [128][128] L2-resident, reused
                      const float* __restrict__ bias,   // [128]
                      float* __restrict__ out)          // [N][128]   written once (NT)
{
    __shared__ float q_lds[2][16][QSTR];   // double-buffered q tile (16 x 64)
    __shared__ float h_lds[16][HSTR];      // diffused features for projection

    const int tid  = threadIdx.x;
    const int lane = tid & 31;
    const int wave = tid >> 5;             // 0..7 -> 16-col output chunk
    const int row0 = blockIdx.x * 16;      // 16-row strip of the output
    const int c    = wave * 16;

    const float th0 = theta[0], th1 = theta[1], th2 = theta[2], th3 = theta[3];

    // q-tile producer mapping: each thread computes 4 consecutive q values
    const int qrow = tid >> 4;             // 0..15
    const int qc4  = (tid & 15) << 2;      // 0,4,...,60 -> float4 per thread

    // WMMA fragment lane mapping (A: lane=M; B/C/D: lane=N; halves via lane[4])
    const int m  = lane & 15;
    const int hi = lane >> 4;

    const size_t slice   = (size_t)Nn * Nn;
    const size_t rowbase = (size_t)(row0 + qrow) * Nn + qc4;

    v8f acc = {0.f, 0.f, 0.f, 0.f, 0.f, 0.f, 0.f, 0.f};

    // ---- prologue: issue streaming (non-temporal) loads for strip 0 ----
    v4f av, t0v, t1v, t2v, t3v;
    av  = __builtin_nontemporal_load((const v4f*)(a + rowbase));
    t0v = __builtin_nontemporal_load((const v4f*)(T + 0 * slice + rowbase));
    t1v = __builtin_nontemporal_load((const v4f*)(T + 1 * slice + rowbase));
    t2v = __builtin_nontemporal_load((const v4f*)(T + 2 * slice + rowbase));
    t3v = __builtin_nontemporal_load((const v4f*)(T + 3 * slice + rowbase));

    for (int jt = 0; jt < Nn / TJ; ++jt) {
        const int j   = jt * TJ;
        const int buf = jt & 1;

        // q = (sum_k theta_k * T_k) * a   for this 16x64 tile
        v4f q = (t0v * th0 + t1v * th1 + t2v * th2 + t3v * th3) * av;
        *(v4f*)&q_lds[buf][qrow][qc4] = q;

        // prefetch next strip before the barrier: HBM stream overlaps WMMA
        if (jt + 1 < Nn / TJ) {
            const size_t nb = rowbase + (size_t)(j + TJ);
            av  = __builtin_nontemporal_load((const v4f*)(a + nb));
            t0v = __builtin_nontemporal_load((const v4f*)(T + 0 * slice + nb));
            t1v = __builtin_nontemporal_load((const v4f*)(T + 1 * slice + nb));
            t2v = __builtin_nontemporal_load((const v4f*)(T + 2 * slice + nb));
            t3v = __builtin_nontemporal_load((const v4f*)(T + 3 * slice + nb));
        }

        __syncthreads();   // single barrier/iter is safe with 2 LDS buffers

        // acc(16x16) += q_tile(16x64) @ x[j:j+64, c:c+16], f32 WMMA K=4 chain
        #pragma unroll
        for (int kk = 0; kk < TJ; kk += 4) {
            const int ka = kk + 2 * hi;
            v2f A, B;
            A.x = q_lds[buf][m][ka];
            A.y = q_lds[buf][m][ka + 1];
            const float* xp = x + (size_t)(j + ka) * DIN + c + m;  // L2-resident (RT hint)
            B.x = xp[0];
            B.y = xp[DIN];
            acc = __builtin_amdgcn_wmma_f32_16x16x4_f32(
                false, A, false, B, (short)0, acc, false, false);
        }
    }

    // ---- PReLU (per-feature slope) and stage h tile in LDS ----
    const float alf = alpha[c + m];
    #pragma unroll
    for (int i = 0; i < 8; ++i) {
        float v = acc[i];
        v = v > 0.0f ? v : alf * v;
        h_lds[i + 8 * hi][c + m] = v;
    }
    __syncthreads();

    // ---- projection: out(16x16 chunk) = h(16x128) @ W^T(:, c:c+16) + b ----
    v8f acc2 = {0.f, 0.f, 0.f, 0.f, 0.f, 0.f, 0.f, 0.f};
    #pragma unroll
    for (int f = 0; f < DIN; f += 4) {
        const int ka = f + 2 * hi;
        v2f A, B;
        A.x = h_lds[m][ka];
        A.y = h_lds[m][ka + 1];
        const float* wp = W + (size_t)(c + m) * DIN + ka;  // B[K][n] = W[c+n][K]
        B.x = wp[0];
        B.y = wp[1];
        acc2 = __builtin_amdgcn_wmma_f32_16x16x4_f32(
            false, A, false, B, (short)0, acc2, false, false);
    }

    const float bb = bias[c + m];
    #pragma unroll
    for (int i = 0; i < 8; ++i) {
        __builtin_nontemporal_store(
            acc2[i] + bb,
            out + (size_t)(row0 + i + 8 * hi) * DOUT + c + m);
    }
}

extern "C" void kernel_launch(void* const* d_in, const int* in_sizes, int n_in,
                              void* d_out, int out_size, void* d_ws, size_t ws_size,
                              hipStream_t stream) {
    (void)in_sizes; (void)n_in; (void)d_ws; (void)ws_size; (void)out_size;
    const float* theta = (const float*)d_in[0];
    const float* T     = (const float*)d_in[1];
    const float* x     = (const float*)d_in[2];
    const float* a     = (const float*)d_in[3];
    const float* alpha = (const float*)d_in[4];
    const float* W     = (const float*)d_in[5];
    const float* bias  = (const float*)d_in[6];
    ggd_fused_kernel<<<dim3(Nn / 16), dim3(256), 0, stream>>>(
        theta, T, x, a, alpha, W, bias, (float*)d_out);
}